// Reccurency_858993459635
// MI455X (gfx1250) — compile-verified
//
#include <hip/hip_runtime.h>
#include <hip/hip_bf16.h>
#include <math.h>

typedef __attribute__((ext_vector_type(2))) float v2f;
typedef __attribute__((ext_vector_type(8))) float v8f;

// ---------------------------------------------------------------------------
// Kernel 1: zero the dist matrix (and ALPHA slot) in d_out, zero sum
// accumulators in d_ws.
// ---------------------------------------------------------------------------
__global__ void k_init(float* __restrict__ out, float* __restrict__ sumbuf,
                       size_t total_plus_alpha, int B) {
    size_t i = (size_t)blockIdx.x * blockDim.x + threadIdx.x;
    if (i < total_plus_alpha) out[i] = 0.0f;   // includes ALPHA slot (== 0.0)
    if (i < (size_t)B) sumbuf[i] = 0.0f;
}

// ---------------------------------------------------------------------------
// Kernel 2: scatter-add contributions. One thread per edge, loop over the
// 64 (h, r) pairs cached in LDS; atomicAdd on (rare) match.
// ---------------------------------------------------------------------------
__global__ void k_scatter(const int* __restrict__ edge_index,
                          const int* __restrict__ edge_type,
                          const float* __restrict__ edge_weight,
                          const int* __restrict__ batch,
                          float* __restrict__ dist,
                          int E, int N, int B) {
    __shared__ int hs[64];
    __shared__ int rs[64];
    for (int i = threadIdx.x; i < B && i < 64; i += blockDim.x) {
        hs[i] = batch[i * 96 + 0];   // batch[b, 0, 0]
        rs[i] = batch[i * 96 + 2];   // batch[b, 0, 2]
    }
    __syncthreads();

    int e = blockIdx.x * blockDim.x + threadIdx.x;
    if (e >= E) return;

    const int   head = edge_index[e];
    const int   tail = edge_index[E + e];
    const int   t    = edge_type[e];
    const float w    = edge_weight[e];

    for (int b = 0; b < B; ++b) {
        if (head == hs[b] && t == rs[b]) {
            atomicAdd(&dist[(size_t)b * N + tail], w);
        }
    }
}

// ---------------------------------------------------------------------------
// Kernel 3: per-row max. One block per row; LDS tree reduction.
// ---------------------------------------------------------------------------
__global__ void k_rowmax(const float* __restrict__ dist,
                         float* __restrict__ maxbuf, int N) {
    __shared__ float sm[256];
    const int b = blockIdx.x;
    const float* rowp = dist + (size_t)b * N;

    float m = -INFINITY;
    for (int i = threadIdx.x; i < N; i += blockDim.x) m = fmaxf(m, rowp[i]);
    sm[threadIdx.x] = m;
    __syncthreads();
    for (int s = 128; s > 0; s >>= 1) {
        if (threadIdx.x < s) sm[threadIdx.x] = fmaxf(sm[threadIdx.x], sm[threadIdx.x + s]);
        __syncthreads();
    }
    if (threadIdx.x == 0) maxbuf[b] = sm[0];
}

// ---------------------------------------------------------------------------
// Kernel 4: per-row sum of exp(x - max) via V_WMMA_F32_16X16X4_F32.
//   blockIdx.y = row group (16 rows). Each wave owns a K-slice; A-matrix is a
//   16x4 f32 tile of exp values, B = ones(4x16), so D accumulates f32 row sums
//   (every column of D identical). Uniform per-wave loop bounds keep EXEC all-1
//   around the WMMA. Lane 0 holds rows 0..7, lane 16 rows 8..15 of the group.
// ---------------------------------------------------------------------------
__global__ void k_rowsumexp(const float* __restrict__ dist,
                            const float* __restrict__ maxbuf,
                            float* __restrict__ sumbuf, int N) {
    const int g    = blockIdx.y;            // 16-row group
    const int tid  = threadIdx.x;           // 0..255
    const int lane = tid & 31;
    const int wavesPerBlock = blockDim.x >> 5;
    const int waveId   = blockIdx.x * wavesPerBlock + (tid >> 5);
    const int numWaves = gridDim.x * wavesPerBlock;

    // A-matrix 16x4 f32 layout: lanes 0-15 -> M=lane, K0/K1; lanes 16-31 -> M=lane-16, K2/K3
    const int row  = lane & 15;
    const int kofs = (lane >> 4) << 1;      // 0 or 2
    const int grow = g * 16 + row;
    const float m  = maxbuf[grow];
    const float* rowp = dist + (size_t)grow * N;

    v2f bmat; bmat.x = 1.0f; bmat.y = 1.0f; // ones(4x16)
    v8f c = {0.f, 0.f, 0.f, 0.f, 0.f, 0.f, 0.f, 0.f};

    const int nChunks = N >> 2;             // K chunks of 4
    for (int ci = waveId; ci < nChunks; ci += numWaves) {
        const int k = (ci << 2) + kofs;
        v2f a;
        a.x = expf(rowp[k]     - m);
        a.y = expf(rowp[k + 1] - m);
        // D = A(16x4) x ones(4x16) + C : accumulates row sums in f32
        c = __builtin_amdgcn_wmma_f32_16x16x4_f32(
                /*neg_a=*/false, a, /*neg_b=*/false, bmat,
                /*c_mod=*/(short)0, c, /*reuse_a=*/false, /*reuse_b=*/false);
    }

    // Extract: D column N=0 lives in lane 0 (M=0..7 -> c[0..7]) and
    // lane 16 (M=8..15 -> c[0..7]). Combine waves via atomics.
    if (lane == 0 || lane == 16) {
        const int rbase = g * 16 + ((lane >> 4) << 3);
#pragma unroll
        for (int j = 0; j < 8; ++j) atomicAdd(&sumbuf[rbase + j], c[j]);
    }

    // Scalar tail for N % 4 (empty for N=50000, kept for generality).
    if (blockIdx.x == 0 && tid < 16) {
        float s = 0.0f;
        for (int k = nChunks << 2; k < N; ++k) s += expf(rowp[k] - m);
        if ((nChunks << 2) < N) atomicAdd(&sumbuf[g * 16 + tid], s);
    }
}

// ---------------------------------------------------------------------------
// Kernel 5: normalize in place: out = exp(x - max) / sum.
// ---------------------------------------------------------------------------
__global__ void k_normalize(float* __restrict__ out,
                            const float* __restrict__ maxbuf,
                            const float* __restrict__ sumbuf,
                            int N, size_t total) {
    size_t i = (size_t)blockIdx.x * blockDim.x + threadIdx.x;
    if (i >= total) return;
    int b = (int)(i / (size_t)N);
    out[i] = expf(out[i] - maxbuf[b]) / sumbuf[b];
}

// ---------------------------------------------------------------------------
// Host launcher. Inputs (setup_inputs order):
//   d_in[0]=edge_index (2,E) int32, d_in[1]=edge_type (E,) int32,
//   d_in[2]=batch (B,32,3) int32, d_in[3]=edge_weight (E,) f32,
//   d_in[4]=num_nodes (device scalar, unused: N derived on host).
// d_out = freq_res (B*N f32) ++ ALPHA (1 f32).
// ---------------------------------------------------------------------------
extern "C" void kernel_launch(void* const* d_in, const int* in_sizes, int n_in,
                              void* d_out, int out_size, void* d_ws, size_t ws_size,
                              hipStream_t stream) {
    const int*   edge_index  = (const int*)d_in[0];
    const int*   edge_type   = (const int*)d_in[1];
    const int*   batch       = (const int*)d_in[2];
    const float* edge_weight = (const float*)d_in[3];

    const int    E     = in_sizes[1];
    const int    B     = in_sizes[2] / 96;          // (B,32,3) -> B
    const size_t total = (size_t)out_size - 1;      // freq_res elements
    const int    N     = (int)(total / (size_t)B);  // num_nodes

    float* out    = (float*)d_out;
    float* maxbuf = (float*)d_ws;                   // B floats
    float* sumbuf = maxbuf + B;                     // B floats

    // 1) zero dist + ALPHA slot + sum accumulators
    {
        size_t n = total + 1;
        int blocks = (int)((n + 255) / 256);
        k_init<<<blocks, 256, 0, stream>>>(out, sumbuf, n, B);
    }
    // 2) scatter-add
    {
        int blocks = (E + 255) / 256;
        k_scatter<<<blocks, 256, 0, stream>>>(edge_index, edge_type, edge_weight,
                                              batch, out, E, N, B);
    }
    // 3) per-row max
    k_rowmax<<<B, 256, 0, stream>>>(out, maxbuf, N);

    // 4) per-row sum of exp via WMMA (B assumed multiple of 16; B=64 here)
    {
        dim3 grid(32, (unsigned)(B / 16));
        k_rowsumexp<<<grid, 256, 0, stream>>>(out, maxbuf, sumbuf, N);
    }
    // 5) normalize
    {
        int blocks = (int)((total + 255) / 256);
        k_normalize<<<blocks, 256, 0, stream>>>(out, maxbuf, sumbuf, N, total);
    }
}